// DeltaNetLayer_65687229825728
// MI455X (gfx1250) — compile-verified
//
#include <hip/hip_runtime.h>
#include <hip/hip_bf16.h>

typedef __attribute__((ext_vector_type(2))) float v2f;
typedef __attribute__((ext_vector_type(8))) float v8f;
typedef __attribute__((ext_vector_type(4))) unsigned int u32x4;
typedef __attribute__((ext_vector_type(8))) int i32x8;
typedef __attribute__((ext_vector_type(4))) int i32x4;

#define LDP 65  // padded row stride (floats) for 64x64 LDS tiles -> conflict-free column reads

__device__ __forceinline__ v8f wmma_f32_4(v2f a, v2f b, v8f c) {
  // D = A(16x4) * B(4x16) + C, full f32 precision (matches f32 reference)
  return __builtin_amdgcn_wmma_f32_16x16x4_f32(false, a, false, b, (short)0, c, false, false);
}

__device__ __forceinline__ unsigned lds_off(const void* p) {
  return (unsigned)(unsigned long long)(uintptr_t)p;  // low 32 bits of generic LDS addr = LDS byte offset
}

// ---------------- TDM: load a 64x64 f32 tile (row stride 1024 floats) into LDS with stride-65 padding
__device__ __forceinline__ void tdm_load_tile(const float* gsrc, unsigned lds_byte_off) {
  unsigned long long ga = (unsigned long long)(uintptr_t)gsrc;
  u32x4 g0;
  g0[0] = 1u;                                            // count=1, user descriptor
  g0[1] = lds_byte_off;                                  // lds_addr
  g0[2] = (unsigned)ga;                                  // global_addr[31:0]
  g0[3] = (unsigned)((ga >> 32) & 0x1FFFFFFu) | (2u << 30); // global_addr[56:32] | type=2
  i32x8 g1;
  // data_size=4B (code 2) | pad_enable | pad_interval=64 DW (code 5) | pad_amount=1 DW (code 0)
  g1[0] = (int)((2u << 16) | (1u << 20) | (5u << 22));
  g1[1] = (int)(64u << 16);   // tensor_dim0[15:0]=64 (bits 63:48)
  g1[2] = (int)(64u << 16);   // tensor_dim0[31:16]=0, tensor_dim1[15:0]=64
  g1[3] = (int)(64u << 16);   // tensor_dim1[31:16]=0, tile_dim0=64
  g1[4] = 64;                 // tile_dim1=64, tile_dim2=0
  g1[5] = 1024;               // tensor_dim0_stride = 1024 elements
  g1[6] = 0;
  g1[7] = 0;
  i32x4 z4 = {0, 0, 0, 0};
  i32x8 z8 = {0, 0, 0, 0, 0, 0, 0, 0};
  __builtin_amdgcn_tensor_load_to_lds(g0, g1, z4, z4, z8, 0);
}

// ---------------- f32 WMMA GEMM: C[MxN] = A[MxK] * B[KxN]; block tile 128x64, 2x2 tiles/wave
__global__ __launch_bounds__(256) void gemm_f32_wmma(const float* __restrict__ A,
                                                     const float* __restrict__ B,
                                                     float* __restrict__ C,
                                                     int M, int N, int K) {
  __shared__ float As[128 * 33];
  __shared__ float Bs[32 * 65];
  const int tid  = threadIdx.x;
  const int wave = tid >> 5;
  const int lane = tid & 31;
  const int half = lane >> 4;
  const int r    = lane & 15;
  const int m0 = blockIdx.y * 128, n0 = blockIdx.x * 64;
  const int ri = (wave >> 1) * 32;   // wave's 32x32 region within block tile
  const int cj = (wave & 1) * 32;
  v8f acc00 = {}, acc01 = {}, acc10 = {}, acc11 = {};
  for (int kb = 0; kb < K; kb += 32) {
    __syncthreads();
    {
      int row = tid >> 5, col = tid & 31;
#pragma unroll
      for (int rr = 0; rr < 16; ++rr)
        As[(row + rr * 8) * 33 + col] = A[(size_t)(m0 + row + rr * 8) * K + kb + col];
      int brow = tid >> 6, bcol = tid & 63;
#pragma unroll
      for (int rr = 0; rr < 8; ++rr)
        Bs[(brow + rr * 4) * 65 + bcol] = B[(size_t)(kb + brow + rr * 4) * N + n0 + bcol];
    }
    __syncthreads();
    v2f a0[8], a1[8], b0[8], b1[8];
#pragma unroll
    for (int kk = 0; kk < 8; ++kk) {
      const int k = kk * 4 + 2 * half;
      a0[kk].x = As[(ri + r) * 33 + k];        a0[kk].y = As[(ri + r) * 33 + k + 1];
      a1[kk].x = As[(ri + 16 + r) * 33 + k];   a1[kk].y = As[(ri + 16 + r) * 33 + k + 1];
      b0[kk].x = Bs[k * 65 + cj + r];          b0[kk].y = Bs[(k + 1) * 65 + cj + r];
      b1[kk].x = Bs[k * 65 + cj + 16 + r];     b1[kk].y = Bs[(k + 1) * 65 + cj + 16 + r];
    }
#pragma unroll
    for (int kk = 0; kk < 8; ++kk) {
      acc00 = wmma_f32_4(a0[kk], b0[kk], acc00);
      acc01 = wmma_f32_4(a0[kk], b1[kk], acc01);
      acc10 = wmma_f32_4(a1[kk], b0[kk], acc10);
      acc11 = wmma_f32_4(a1[kk], b1[kk], acc11);
    }
  }
#pragma unroll
  for (int p = 0; p < 8; ++p) {
    const int mr0 = m0 + ri + p + 8 * half;
    C[(size_t)mr0 * N + n0 + cj + r]           = acc00[p];
    C[(size_t)mr0 * N + n0 + cj + 16 + r]      = acc01[p];
    C[(size_t)(mr0 + 16) * N + n0 + cj + r]      = acc10[p];
    C[(size_t)(mr0 + 16) * N + n0 + cj + 16 + r] = acc11[p];
  }
}

// ---------------- beta = sigmoid(x @ Wbeta + b), layout [8192][16]
__global__ __launch_bounds__(256) void beta_kernel(const float* __restrict__ x,
                                                   const float* __restrict__ Wb,
                                                   const float* __restrict__ bb,
                                                   float* __restrict__ beta, int TT) {
  int gid = blockIdx.x * 256 + threadIdx.x;
  int t = gid >> 4, h = gid & 15;
  if (t >= TT) return;
  float s = bb[h];
  const float* xr = x + (size_t)t * 1024;
  for (int i = 0; i < 1024; ++i) s += xr[i] * Wb[i * 16 + h];
  beta[gid] = 1.f / (1.f + __expf(-s));
}

// ---------------- L2-normalize each 64-float key segment (16 lanes per row, float4 each)
__global__ __launch_bounds__(256) void knorm_kernel(float* __restrict__ Kb, int rows) {
  int tid = threadIdx.x;
  int grp = (blockIdx.x * 256 + tid) >> 4;
  int l = tid & 15;
  if (grp >= rows) return;
  float4* p = (float4*)Kb + (size_t)grp * 16 + l;
  float4 v = *p;
  float ss = v.x * v.x + v.y * v.y + v.z * v.z + v.w * v.w;
#pragma unroll
  for (int m = 8; m >= 1; m >>= 1) ss += __shfl_xor(ss, m, 32);
  float sc = 1.f / fmaxf(sqrtf(ss), 1e-12f);
  v.x *= sc; v.y *= sc; v.z *= sc; v.w *= sc;
  *p = v;
}

// ---------------- 64x64x64 workgroup matmul on WMMA f32 16x16x4; LDS operands, stride LDP
// wave handles tiles (i0, j16) and (i0+32, j16): B fragment shared, loads batched 8 k-steps deep
__device__ __forceinline__ void mm64(const float* A, bool at, const float* B, bool bt,
                                     const float* C, float* D, int tid) {
  __syncthreads();
  const int wave = tid >> 5, lane = tid & 31, half = lane >> 4, r = lane & 15;
  const int i0 = (wave >> 2) * 16, i1 = i0 + 32, j16 = (wave & 3) * 16;
  v8f acc0 = {}, acc1 = {};
  if (C) {
#pragma unroll
    for (int p = 0; p < 8; ++p) {
      acc0[p] = C[(i0 + p + 8 * half) * LDP + j16 + r];
      acc1[p] = C[(i1 + p + 8 * half) * LDP + j16 + r];
    }
  }
#pragma unroll
  for (int kb = 0; kb < 2; ++kb) {
    v2f a0[8], a1[8], b[8];
#pragma unroll
    for (int kk = 0; kk < 8; ++kk) {
      const int k = kb * 32 + kk * 4 + 2 * half;
      if (at) {
        a0[kk].x = A[k * LDP + i0 + r];       a0[kk].y = A[(k + 1) * LDP + i0 + r];
        a1[kk].x = A[k * LDP + i1 + r];       a1[kk].y = A[(k + 1) * LDP + i1 + r];
      } else {
        a0[kk].x = A[(i0 + r) * LDP + k];     a0[kk].y = A[(i0 + r) * LDP + k + 1];
        a1[kk].x = A[(i1 + r) * LDP + k];     a1[kk].y = A[(i1 + r) * LDP + k + 1];
      }
      if (bt) {
        b[kk].x = B[(j16 + r) * LDP + k];     b[kk].y = B[(j16 + r) * LDP + k + 1];
      } else {
        b[kk].x = B[k * LDP + j16 + r];       b[kk].y = B[(k + 1) * LDP + j16 + r];
      }
    }
#pragma unroll
    for (int kk = 0; kk < 8; ++kk) {
      acc0 = wmma_f32_4(a0[kk], b[kk], acc0);
      acc1 = wmma_f32_4(a1[kk], b[kk], acc1);
    }
  }
#pragma unroll
  for (int p = 0; p < 8; ++p) {
    D[(i0 + p + 8 * half) * LDP + j16 + r] = acc0[p];
    D[(i1 + p + 8 * half) * LDP + j16 + r] = acc1[p];
  }
  __syncthreads();
}

// ---------------- chunked DeltaNet scan: 1 workgroup per (b,h); 32 chunks of 64 steps
__global__ __launch_bounds__(256) void deltanet_scan(const float* __restrict__ Qb,
                                                     const float* __restrict__ Kb,
                                                     float* __restrict__ Vb,   // O overwrites V
                                                     const float* __restrict__ betab) {
  __shared__ float S_T[64 * LDP], Kc[64 * LDP], Vc[64 * LDP], Qc[64 * LDP];
  __shared__ float U[64 * LDP], Am[64 * LDP], T1[64 * LDP], T2[64 * LDP];
  __shared__ float betar[64];
  const int tid = threadIdx.x;
  const int bh = blockIdx.x;
  const int b = bh >> 4, h = bh & 15;
  const size_t rowbase = (size_t)b * 2048;

  for (int i = tid; i < 64 * LDP; i += 256) S_T[i] = 0.f;
  __syncthreads();

  for (int c = 0; c < 32; ++c) {
    const int t0 = c * 64;
    const size_t goff = (rowbase + t0) * 1024 + (size_t)h * 64;
    if (tid == 0) {  // one wave issues the three TDM tile loads, then drains TENSORcnt
      tdm_load_tile(Qb + goff, lds_off(Qc));
      tdm_load_tile(Kb + goff, lds_off(Kc));
      tdm_load_tile(Vb + goff, lds_off(Vc));
      __builtin_amdgcn_s_wait_tensorcnt(0);
    }
    if (tid < 64) betar[tid] = betab[(rowbase + t0 + tid) * 16 + h];
    __syncthreads();

    // G = K K^T ; Am = beta .* strict_tril(G)
    mm64(Kc, false, Kc, true, nullptr, T1, tid);
    for (int i = tid; i < 4096; i += 256) {
      int t = i >> 6, s = i & 63;
      Am[t * LDP + s] = (s < t) ? betar[t] * T1[t * LDP + s] : 0.f;
    }
    // U0 = beta .* (V - K S^T)
    mm64(Kc, false, S_T, false, nullptr, T1, tid);
    for (int i = tid; i < 4096; i += 256) {
      int t = i >> 6, s = i & 63;
      U[t * LDP + s] = betar[t] * (Vc[t * LDP + s] - T1[t * LDP + s]);
    }
    // U = (I+Am)^-1 U0 via nilpotent factorization: apply (I - Am^(2^k)), k=0..5
    {
      float* P = Am; float* Pn = T2;
      for (int iter = 0; iter < 6; ++iter) {
        mm64(P, false, U, false, nullptr, T1, tid);
        for (int i = tid; i < 4096; i += 256) {
          int t = i >> 6, s = i & 63;
          U[t * LDP + s] -= T1[t * LDP + s];
        }
        if (iter < 5) {
          mm64(P, false, P, false, nullptr, Pn, tid);
          float* tmp = P; P = Pn; Pn = tmp;
        }
      }
    }
    // O = Q S^T + tril_incl(Q K^T) U
    mm64(Qc, false, S_T, false, nullptr, T1, tid);      // T1 = Q S^T
    mm64(Qc, false, Kc, true, nullptr, Am, tid);        // Am = Q K^T
    for (int i = tid; i < 4096; i += 256) {
      int t = i >> 6, s = i & 63;
      if (s > t) Am[t * LDP + s] = 0.f;
    }
    mm64(Am, false, U, false, T1, T2, tid);             // T2 = O chunk
    for (int i = tid; i < 4096; i += 256) {
      int t = i >> 6, s = i & 63;
      Vb[(rowbase + t0 + t) * 1024 + (size_t)h * 64 + s] = T2[t * LDP + s];
    }
    // S^T += K^T U
    mm64(Kc, true, U, false, S_T, S_T, tid);
  }
}

extern "C" void kernel_launch(void* const* d_in, const int* in_sizes, int n_in,
                              void* d_out, int out_size, void* d_ws, size_t ws_size,
                              hipStream_t stream) {
  const float* x     = (const float*)d_in[0];
  const float* Wq    = (const float*)d_in[1];
  const float* Wk    = (const float*)d_in[2];
  const float* Wv    = (const float*)d_in[3];
  const float* Wbeta = (const float*)d_in[4];
  const float* bbeta = (const float*)d_in[5];
  const float* Wout  = (const float*)d_in[6];
  float* out = (float*)d_out;
  float* ws = (float*)d_ws;
  const size_t MATF = 8192ull * 1024ull;  // 8.4M floats
  float* Qb    = ws;
  float* Kb    = ws + MATF;
  float* Vb    = ws + 2 * MATF;
  float* betab = ws + 3 * MATF;

  dim3 g(16, 64), blk(256);
  gemm_f32_wmma<<<g, blk, 0, stream>>>(x, Wq, Qb, 8192, 1024, 1024);
  gemm_f32_wmma<<<g, blk, 0, stream>>>(x, Wk, Kb, 8192, 1024, 1024);
  gemm_f32_wmma<<<g, blk, 0, stream>>>(x, Wv, Vb, 8192, 1024, 1024);
  beta_kernel<<<512, 256, 0, stream>>>(x, Wbeta, bbeta, betab, 8192);
  knorm_kernel<<<8192, 256, 0, stream>>>(Kb, 131072);
  deltanet_scan<<<64, 256, 0, stream>>>(Qb, Kb, Vb, betab);
  gemm_f32_wmma<<<g, blk, 0, stream>>>(Vb, Wout, out, 8192, 1024, 1024);
}